// Class_PixelNCELoss_2199023255631
// MI455X (gfx1250) — compile-verified
//
#include <hip/hip_runtime.h>

// PixelNCE contrastive loss for MI455X (gfx1250).
// Shapes fixed by the reference: B=4, C=256, H*W=16384, nc=19, nv=50, P=950.
// sim GEMM runs on V_WMMA_F32_16X16X4_F32 (fp32 matrix path, matches reference
// precision); P padded to 960 (60 tiles of 16).

typedef __attribute__((ext_vector_type(2))) float v2f;
typedef __attribute__((ext_vector_type(8))) float v8f;

#define B_    4
#define C_    256
#define HW_   16384
#define NV_   50
#define P_    950
#define PP_   960            // P padded to multiple of 16
#define PPAD  968            // LDS row stride: 968 % 64 == 8 -> conflict-free scan
#define NTILE (PP_ / 16)     // 60 column tiles
#define TINV  (1.0f / 0.07f)

// ---------------------------------------------------------------------------
// Kernel 1: gather anchors from channel-major feats into row-major [b][p][c],
// zero-padding rows p in [950, 960). Whole input (134 MB) fits in 192 MB L2,
// so cacheline reuse across p happens in L2; HBM traffic ~= unique lines.
// grid = (PP_, B_), block = C_ threads (one per channel).
// ---------------------------------------------------------------------------
__global__ void __launch_bounds__(C_)
gather_kernel(const float* __restrict__ fq, const float* __restrict__ fk,
              const int* __restrict__ sidx,
              float* __restrict__ Xq, float* __restrict__ Xk) {
    const int p = blockIdx.x;
    const int b = blockIdx.y;
    const int c = threadIdx.x;
    float vq = 0.0f, vk = 0.0f;
    if (p < P_) {
        const int idx = sidx[b * P_ + p];                 // pixel index in [0, HW)
        const size_t off = (size_t)(b * C_ + c) * HW_ + (size_t)idx;
        vq = fq[off];
        vk = fk[off];
    }
    const size_t o = (size_t)(b * PP_ + p) * C_ + c;
    Xq[o] = vq;
    Xk[o] = vk;
}

// ---------------------------------------------------------------------------
// Kernel 2: per (batch, 16-row tile) block. 4 waves (128 threads) compute the
// full 16 x 960 strip of sim via fp32 WMMA into LDS, then all 128 threads do
// the masked logsumexp: 8 threads per row + shfl_xor butterfly combine.
//
// WMMA f32 16x16x4 per-lane layout (ISA 7.12.2):
//   A (16x4):  lane L<16 -> M=L, VGPR{0,1}=K{0,1}; lane>=16 -> M=L-16, K{2,3}
//   B (4x16):  lane L<16 -> N=L, VGPR{0,1}=K{0,1}; lane>=16 -> N=L-16, K{2,3}
//   D (16x16): VGPR r, lanes 0-15 -> M=r, lanes 16-31 -> M=r+8; N = lane&15
// Both A and B reduce to one contiguous 64-bit load per lane per K-step.
// ---------------------------------------------------------------------------
__global__ void __launch_bounds__(128)
nce_kernel(const float* __restrict__ Xq, const float* __restrict__ Xk,
           float* __restrict__ out) {
    __shared__ float simbuf[16 * PPAD];   // 61952 B

    const int b       = blockIdx.y;
    const int rowbase = blockIdx.x * 16;
    const int lane    = threadIdx.x & 31;
    const int wave    = threadIdx.x >> 5;
    const int laneLo  = lane & 15;
    const int laneHi  = lane >> 4;        // 0 or 1

    // per-lane base pointer into the A row-block (shared by all column tiles)
    const float* arow = Xq + (size_t)(b * PP_ + rowbase + laneLo) * C_ + laneHi * 2;

    for (int j = wave; j < NTILE; j += 4) {          // uniform per wave -> EXEC all 1s
        const int colbase = j * 16;
        const float* brow =
            Xk + (size_t)(b * PP_ + colbase + laneLo) * C_ + laneHi * 2;

        v8f acc = {0.f, 0.f, 0.f, 0.f, 0.f, 0.f, 0.f, 0.f};
#pragma unroll 8
        for (int k0 = 0; k0 < C_; k0 += 4) {
            v2f av = *(const v2f*)(arow + k0);
            v2f bv = *(const v2f*)(brow + k0);
            // D = A x B + C on the fp32 matrix pipe
            acc = __builtin_amdgcn_wmma_f32_16x16x4_f32(
                /*neg_a=*/false, av, /*neg_b=*/false, bv,
                /*c_mod=*/(short)0, acc, /*reuse_a=*/false, /*reuse_b=*/false);
        }

        // spill D tile to the LDS strip (consecutive lanes -> consecutive cols)
        const int col   = colbase + laneLo;
        const int rbase = laneHi * 8;
#pragma unroll
        for (int r = 0; r < 8; ++r)
            simbuf[(rbase + r) * PPAD + col] = acc[r];
    }

    __syncthreads();

    // Masked logsumexp per row. Softmax set = {sim[p,p]} U {sim[p,q] : q's
    // 50-wide block != p's}; loss = logsumexp(set/T) - sim[p,p]/T.
    // 8 threads per row, strided scan, butterfly combine within the 8-group.
    {
        const int m   = threadIdx.x >> 3;     // row 0..15
        const int sub = threadIdx.x & 7;      // 0..7 within row group
        const int p   = rowbase + m;
        if (p < P_) {
            const float* row = simbuf + m * PPAD;
            const int lo = (p / NV_) * NV_;   // same-class block [lo, lo+NV)
            const int hi = lo + NV_;

            float mx = -3.4e38f;
            for (int q = sub; q < P_; q += 8) {
                if (q < lo || q >= hi || q == p)
                    mx = fmaxf(mx, row[q] * TINV);
            }
#pragma unroll
            for (int o = 1; o < 8; o <<= 1)
                mx = fmaxf(mx, __shfl_xor(mx, o, 8));

            float s = 0.0f;
            for (int q = sub; q < P_; q += 8) {
                if (q < lo || q >= hi || q == p)
                    s += expf(row[q] * TINV - mx);
            }
#pragma unroll
            for (int o = 1; o < 8; o <<= 1)
                s += __shfl_xor(s, o, 8);

            if (sub == 0)
                out[b * P_ + p] = (mx + logf(s)) - row[p] * TINV;
        }
    }
}

// ---------------------------------------------------------------------------
extern "C" void kernel_launch(void* const* d_in, const int* in_sizes, int n_in,
                              void* d_out, int out_size, void* d_ws, size_t ws_size,
                              hipStream_t stream) {
    const float* fq   = (const float*)d_in[0];   // feats_q [4,256,128,128] f32
    const float* fk   = (const float*)d_in[1];   // feats_k [4,256,128,128] f32
    const int*   sidx = (const int*)d_in[2];     // sample_idx [4,19,50] i32
    float*       out  = (float*)d_out;           // [3800] f32

    float* Xq = (float*)d_ws;                        // [4][960][256]
    float* Xk = Xq + (size_t)B_ * PP_ * C_;          // [4][960][256]

    dim3 g1(PP_, B_);
    gather_kernel<<<g1, C_, 0, stream>>>(fq, fk, sidx, Xq, Xk);

    dim3 g2(NTILE, B_);
    nce_kernel<<<g2, 128, 0, stream>>>(Xq, Xk, out);
}